// LowRankRNN_58145267253493
// MI455X (gfx1250) — compile-verified
//
#include <hip/hip_runtime.h>
#include <math.h>

// Problem dims (fixed by the reference)
#define BATCH 64
#define TLEN  1024
#define IDIM  8
#define HDIM  512
#define ODIM  2

typedef __attribute__((ext_vector_type(2))) float v2f;
typedef __attribute__((ext_vector_type(8))) float v8f;

// ---------------------------------------------------------------------------
// Kernel 1: inp_proj = input(B*T, 8) @ (wi*si)(8, 512)  via V_WMMA_F32_16X16X4_F32
// One wave per 16x16 output tile; K=8 split into two accumulating K=4 WMMAs.
// Result written into the traj region of d_out (scan kernel consumes then
// overwrites it in place).
// ---------------------------------------------------------------------------
__global__ __launch_bounds__(256) void lrnn_inproj_wmma(
    const float* __restrict__ X,   // (B*T, IDIM) row-major
    const float* __restrict__ wi,  // (IDIM, HDIM) row-major
    const float* __restrict__ si,  // (IDIM)
    float* __restrict__ P)         // (B*T, HDIM)
{
    const int NT = HDIM / 16;                 // 32 N-tiles
    const int waveInBlock = threadIdx.x >> 5;
    const int lane        = threadIdx.x & 31;
    const int tile  = blockIdx.x * 8 + waveInBlock;   // wave-tile id
    const int mt    = tile >> 5;              // / NT
    const int nt    = tile & 31;              // % NT
    const int mbase = mt << 4;
    const int nbase = nt << 4;

    const int halfsel = lane >> 4;            // 0: lanes 0-15, 1: lanes 16-31
    const int l15     = lane & 15;

    // A operand (16x4 f32): lane holds row (mbase+l15), K = 2*halfsel + {0,1}
    const float* xrow = X + (size_t)(mbase + l15) * IDIM;
    v2f a0, a1;
    a0.x = xrow[2 * halfsel + 0];             // K-chunk 0: k = 0..3
    a0.y = xrow[2 * halfsel + 1];
    a1.x = xrow[4 + 2 * halfsel + 0];         // K-chunk 1: k = 4..7
    a1.y = xrow[4 + 2 * halfsel + 1];

    // B operand (4x16 f32): VGPR0 = row K=2*halfsel, VGPR1 = row K=2*halfsel+1
    const int n = nbase + l15;
    v2f b0, b1;
    {
        int k = 2 * halfsel;
        b0.x = wi[(k + 0) * HDIM + n] * si[k + 0];
        b0.y = wi[(k + 1) * HDIM + n] * si[k + 1];
        b1.x = wi[(4 + k + 0) * HDIM + n] * si[4 + k + 0];
        b1.y = wi[(4 + k + 1) * HDIM + n] * si[4 + k + 1];
    }

    v8f c = {};
    // 8 args: (neg_a, A, neg_b, B, c_mod, C, reuse_a, reuse_b)
    c = __builtin_amdgcn_wmma_f32_16x16x4_f32(false, a0, false, b0, (short)0, c, false, false);
    c = __builtin_amdgcn_wmma_f32_16x16x4_f32(false, a1, false, b1, (short)0, c, false, false);

    // D (16x16 f32): VGPR r -> row (mbase + r + 8*halfsel), col n
    const int rowoff = mbase + 8 * halfsel;
#pragma unroll
    for (int r = 0; r < 8; ++r)
        P[(size_t)(rowoff + r) * HDIM + n] = c[r];
}

// ---------------------------------------------------------------------------
// Kernel 2: sequential scan. One block per batch element, thread h owns state
// element h. One fused 4-value reduction (r.n0, r.n1, r.wo0, r.wo1) per step:
// components 0,1 feed step t+1's recurrent drive, components 2,3 are out_t.
// ---------------------------------------------------------------------------
__global__ __launch_bounds__(HDIM) void lrnn_scan(
    const float* __restrict__ noise, // (B, T, H)
    const float* __restrict__ m,     // (H, 2)
    const float* __restrict__ nmat,  // (H, 2)
    const float* __restrict__ wo,    // (H, 2)
    const float* __restrict__ so,    // (2)
    const float* __restrict__ h0,    // (H)
    float* __restrict__ out,         // (B, T, 2)
    float* __restrict__ traj)        // (B, T, H), pre-filled with inp_proj
{
    const int b    = blockIdx.x;
    const int h    = threadIdx.x;
    const int wave = h >> 5;
    const int lane = h & 31;
    const float invH = 1.0f / (float)HDIM;

    __shared__ float4 partial[HDIM / 32];   // 16 wave partials

    // Per-thread constants
    const float nn0 = nmat[h * 2 + 0];
    const float nn1 = nmat[h * 2 + 1];
    const float mm0 = m[h * 2 + 0];
    const float mm1 = m[h * 2 + 1];
    const float w0  = wo[h * 2 + 0] * so[0];
    const float w1  = wo[h * 2 + 1] * so[1];

    float hv = h0[h];
    float rv = tanhf(hv);

    const float* nptr = noise + (size_t)b * TLEN * HDIM + h;
    float*       tptr = traj  + (size_t)b * TLEN * HDIM + h;
    float*       optr = out   + (size_t)b * TLEN * ODIM;

    // Pre-loop reduction: q for t = 0 (output components unused)
    float q0, q1;
    {
        float v0 = rv * nn0, v1 = rv * nn1;
#pragma unroll
        for (int off = 16; off; off >>= 1) {
            v0 += __shfl_xor(v0, off, 32);
            v1 += __shfl_xor(v1, off, 32);
        }
        if (lane == 0) partial[wave] = make_float4(v0, v1, 0.0f, 0.0f);
        __syncthreads();
        float s0 = 0.0f, s1 = 0.0f;
#pragma unroll
        for (int w = 0; w < HDIM / 32; ++w) { s0 += partial[w].x; s1 += partial[w].y; }
        q0 = s0; q1 = s1;
        __syncthreads();
    }

    for (int t = 0; t < TLEN; ++t) {
        const float nz = *nptr;     // noise[b,t,h]
        const float p  = *tptr;     // inp_proj[b,t,h] (read before overwrite)
        if (t + 1 < TLEN) {         // hide next step's load latency
            __builtin_prefetch(nptr + HDIM, 0, 3);
            __builtin_prefetch(tptr + HDIM, 0, 3);
        }

        const float rec = (mm0 * q0 + mm1 * q1) * invH;
        hv = hv + 0.05f * nz + 0.2f * (-hv + rec + p);
        rv = tanhf(hv);
        *tptr = hv;                 // traj[b,t,h] = h

        // Fused reduction over H: {r.n0, r.n1, r.wo0, r.wo1}
        float v0 = rv * nn0, v1 = rv * nn1, v2 = rv * w0, v3 = rv * w1;
#pragma unroll
        for (int off = 16; off; off >>= 1) {
            v0 += __shfl_xor(v0, off, 32);
            v1 += __shfl_xor(v1, off, 32);
            v2 += __shfl_xor(v2, off, 32);
            v3 += __shfl_xor(v3, off, 32);
        }
        if (lane == 0) partial[wave] = make_float4(v0, v1, v2, v3);
        __syncthreads();
        float s0 = 0.0f, s1 = 0.0f, s2 = 0.0f, s3 = 0.0f;
#pragma unroll
        for (int w = 0; w < HDIM / 32; ++w) {
            const float4 pv = partial[w];
            s0 += pv.x; s1 += pv.y; s2 += pv.z; s3 += pv.w;
        }
        q0 = s0; q1 = s1;           // recurrent drive for step t+1
        if (h == 0) {
            optr[t * ODIM + 0] = s2 * invH;
            optr[t * ODIM + 1] = s3 * invH;
        }
        __syncthreads();

        nptr += HDIM;
        tptr += HDIM;
    }
}

// ---------------------------------------------------------------------------
// Host entry
// ---------------------------------------------------------------------------
extern "C" void kernel_launch(void* const* d_in, const int* in_sizes, int n_in,
                              void* d_out, int out_size, void* d_ws, size_t ws_size,
                              hipStream_t stream) {
    const float* input = (const float*)d_in[0];  // (B,T,I)
    const float* noise = (const float*)d_in[1];  // (B,T,H)
    const float* wi    = (const float*)d_in[2];  // (I,H)
    const float* si    = (const float*)d_in[3];  // (I)
    const float* m     = (const float*)d_in[4];  // (H,R)
    const float* nmat  = (const float*)d_in[5];  // (H,R)
    const float* wo    = (const float*)d_in[6];  // (H,O)
    const float* so    = (const float*)d_in[7];  // (O)
    const float* h0    = (const float*)d_in[8];  // (H)

    float* out  = (float*)d_out;                         // (B,T,O) first
    float* traj = out + (size_t)BATCH * TLEN * ODIM;     // (B,T,H) second

    // Phase 1: inp_proj -> traj region (WMMA).
    // tiles = (B*T/16) * (H/16) = 4096 * 32 = 131072; 8 waves per block.
    const int tiles  = (BATCH * TLEN / 16) * (HDIM / 16);
    const int blocks = tiles / 8;
    lrnn_inproj_wmma<<<blocks, 256, 0, stream>>>(input, wi, si, traj);

    // Phase 2: sequential scan, one block per batch element.
    lrnn_scan<<<BATCH, HDIM, 0, stream>>>(noise, m, nmat, wo, so, h0, out, traj);
}